// HigherPointV3_6846177869888
// MI455X (gfx1250) — compile-verified
//
#include <hip/hip_runtime.h>
#include <hip/hip_bf16.h>
#include <stdint.h>

typedef __attribute__((ext_vector_type(2))) float v2f;
typedef __attribute__((ext_vector_type(8))) float v8f;
typedef __attribute__((ext_vector_type(4))) unsigned int v4u;
typedef __attribute__((ext_vector_type(4))) int v4i;
typedef __attribute__((ext_vector_type(8))) int v8i;

// dims
#define PP   64
#define CIN  16
#define COUT 16
#define TT   4
#define NN   32
#define XX   64

// derived strides (floats)
#define F_PSTRIDE (CIN * NN * NN * XX)   // 1048576
#define F_ZSTRIDE (NN * NN * XX)         // 65536
#define PF_PSTRIDE (TT * NN * XX)        // 8192

// Issue a TDM DMA: load a 2D tile (tile_dim0=64 f32 per row, tile_dim1=16 rows,
// row stride = F_ZSTRIDE elements) from global into LDS at lds_byte_addr.
// D# group0: count=1 | lds_addr | global_addr[56:0] | type=2
// D# group1: data_size=4B, tensor_dim0=64, tensor_dim1=16, tile=64x16,
//            tensor_dim0_stride=65536
__device__ __forceinline__ void tdm_load_f_tile(uint32_t lds_byte_addr,
                                                uint64_t gaddr_bytes) {
    v4u g0;
    g0.x = 1u;  // count=1, user descriptor, gather off
    g0.y = (unsigned)__builtin_amdgcn_readfirstlane((int)lds_byte_addr);
    g0.z = (unsigned)__builtin_amdgcn_readfirstlane((int)(uint32_t)gaddr_bytes);
    g0.w = ((unsigned)__builtin_amdgcn_readfirstlane(
                (int)(uint32_t)(gaddr_bytes >> 32)) & 0x01FFFFFFu) | (2u << 30);
    v8i g1;
    g1[0] = 0x00020000;        // workgroup_mask=0, data_size=2 (4 bytes)
    g1[1] = (int)(64u << 16);  // tensor_dim0 = 64 (low 16 in [31:16])
    g1[2] = (int)(16u << 16);  // tensor_dim0 hi=0 | tensor_dim1 = 16 (low 16)
    g1[3] = (int)(64u << 16);  // tensor_dim1 hi=0 | tile_dim0 = 64
    g1[4] = 16;                // tile_dim1 = 16, tile_dim2 = 0
    g1[5] = F_ZSTRIDE;         // tensor_dim0_stride low 32 (65536 elements)
    g1[6] = 0;                 // stride hi, tensor_dim1_stride lo (unused, 2D)
    g1[7] = 0;
    v4i gz4 = {0, 0, 0, 0};            // groups 2/3 unused (2D tensor)
    v8i gz8 = {0, 0, 0, 0, 0, 0, 0, 0};
    __builtin_amdgcn_tensor_load_to_lds(g0, g1, gz4, gz4, gz8, 0);
}

__global__ __launch_bounds__(64) void higher_point_kernel(
    const float* __restrict__ F,    // (P, CIN, N, N, X)
    const float* __restrict__ PF,   // (P, T, N, X)
    const float* __restrict__ W0,   // (T, CIN, COUT)
    const float* __restrict__ W1,   // (T, CIN, COUT)
    const float* __restrict__ BIAS, // (COUT,1,1)
    float* __restrict__ OUT)        // (P, COUT, N, N)
{
    __shared__ __align__(16) float sW0[CIN * TT * COUT];  // [z*64+a*16+v] 4KB
    __shared__ __align__(16) float sW1[CIN * TT * COUT];  // [z*64+b*16+v] 4KB
    __shared__ __align__(16) float sPF[PF_PSTRIDE];       // [t*2048+n*64+x] 32KB
    __shared__ __align__(16) float sF[2][2][CIN * XX];    // [wave][buf] 16KB
    __shared__ __align__(16) float sH[2][CIN * 16];       // [wave][z*16+q] 2KB

    const int tid  = threadIdx.x;
    const int wave = tid >> 5;
    const int lane = tid & 31;
    const int i    = blockIdx.x;   // 0..31
    const int p    = blockIdx.y;   // 0..63

    // ---- stage re-layouted weights: sW0[z,a,v]=W0[a,z,v], sW1[z,b,v]=W1[b,z,v]
    for (int t = tid; t < CIN * TT * COUT; t += 64) {
        const int v = t & 15;
        const int a = (t >> 4) & 3;
        const int z = t >> 6;
        sW0[t] = W0[(a * CIN + z) * COUT + v];
        sW1[t] = W1[(a * CIN + z) * COUT + v];
    }
    // ---- stage pair_features block for this p (contiguous 32KB) ----
    {
        const float4* src = (const float4*)(PF + (size_t)p * PF_PSTRIDE);
        float4* dst = (float4*)sPF;
        for (int t = tid; t < PF_PSTRIDE / 4; t += 64) dst[t] = src[t];
    }
    __syncthreads();

    const float bias_v = (lane < COUT) ? BIAS[lane] : 0.0f;

    // lane roles for WMMA fragments
    const int col = lane & 15;            // A-frag: row z; B-frag: col q=(a,b)
    const int lh  = (lane >> 4) << 1;     // K sub-offset: 0 / 2 per half-wave
    const int a_idx = col >> 2;
    const int b_idx = col & 3;

    const float* Arow = &sPF[a_idx * (NN * XX) + i * XX];  // A_pf[a, i, :]
    const size_t fbase = (size_t)p * F_PSTRIDE + (size_t)i * (NN * XX);
    const float* Fsrc0 = F + fbase + (size_t)(wave * 16) * XX;

    // ---- prologue: async-DMA the first F tile for this wave ----
    tdm_load_f_tile((uint32_t)(uintptr_t)&sF[wave][0][0], (uint64_t)(uintptr_t)Fsrc0);

    // each wave handles 16 consecutive j, double-buffered via TDM
    for (int t = 0; t < 16; ++t) {
        const int j = wave * 16 + t;

        if (t + 1 < 16) {
            // prefetch next tile into the other buffer, wait for current
            tdm_load_f_tile((uint32_t)(uintptr_t)&sF[wave][(t + 1) & 1][0],
                            (uint64_t)(uintptr_t)(F + fbase + (size_t)(j + 1) * XX));
            __builtin_amdgcn_s_wait_tensorcnt(1);
        } else {
            __builtin_amdgcn_s_wait_tensorcnt(0);
        }

        // ---- H[z,q] = F_slice (16x64) @ AB (64x16): 16 x v_wmma_f32_16x16x4_f32
        const float* Frow = &sF[wave][t & 1][col * XX];        // F[z=col, :]
        const float* Brow = &sPF[b_idx * (NN * XX) + j * XX];  // B_pf[b, j, :]
        v8f acc = {0.f, 0.f, 0.f, 0.f, 0.f, 0.f, 0.f, 0.f};
#pragma unroll
        for (int k = 0; k < 16; ++k) {
            const int x0 = (k << 2) + lh;
            v2f af = *(const v2f*)(Frow + x0);   // A-frag
            v2f av = *(const v2f*)(Arow + x0);
            v2f bv = *(const v2f*)(Brow + x0);
            v2f bf = av * bv;                    // B-frag: AB[x0..x0+1, q]
            acc = __builtin_amdgcn_wmma_f32_16x16x4_f32(
                false, af, false, bf, (short)0, acc, false, false);
        }

        // ---- spill H (D-layout: VGPR r -> row r / r+8; lane%16 -> col) ----
        {
            const int rbase = (lane < 16) ? 0 : 8;
#pragma unroll
            for (int r = 0; r < 8; ++r)
                sH[wave][(r + rbase) * 16 + col] = acc[r];
        }
        __syncthreads();

        // ---- out[v] = relu( sum_{z,a} W0[a,z,v] * sum_b H[z,a4+b]*W1[b,z,v] + bias )
        if (lane < COUT) {
            const float* Hb = sH[wave];
            float s = bias_v;
#pragma unroll 4
            for (int z = 0; z < CIN; ++z) {
                const float* Hz  = Hb + z * 16;
                const float* W0z = sW0 + z * 64 + lane;
                const float* W1z = sW1 + z * 64 + lane;
#pragma unroll
                for (int a = 0; a < 4; ++a) {
                    float tmp = Hz[4 * a] * W1z[0];
                    tmp = fmaf(Hz[4 * a + 1], W1z[16], tmp);
                    tmp = fmaf(Hz[4 * a + 2], W1z[32], tmp);
                    tmp = fmaf(Hz[4 * a + 3], W1z[48], tmp);
                    s = fmaf(tmp, W0z[a * 16], s);
                }
            }
            s = fmaxf(s, 0.0f);
            OUT[((size_t)(p * COUT + lane) * NN + i) * NN + j] = s;
        }
    }
}

extern "C" void kernel_launch(void* const* d_in, const int* in_sizes, int n_in,
                              void* d_out, int out_size, void* d_ws, size_t ws_size,
                              hipStream_t stream) {
    (void)in_sizes; (void)n_in; (void)out_size; (void)d_ws; (void)ws_size;
    const float* F    = (const float*)d_in[0];
    const float* PF   = (const float*)d_in[1];
    const float* W0   = (const float*)d_in[2];
    const float* W1   = (const float*)d_in[3];
    const float* BIAS = (const float*)d_in[4];
    float* OUT = (float*)d_out;

    dim3 grid(NN, PP);   // (i, p)
    dim3 block(64);      // 2 wave32
    higher_point_kernel<<<grid, block, 0, stream>>>(F, PF, W0, W1, BIAS, OUT);
}